// D3CGTrainer_cond_56745107915493
// MI455X (gfx1250) — compile-verified
//
#include <hip/hip_runtime.h>

typedef float v2f __attribute__((ext_vector_type(2)));
typedef float v4f __attribute__((ext_vector_type(4)));
typedef float v8f __attribute__((ext_vector_type(8)));

#define T_STEPS 1000
#define BETA_1f 1.0e-4f
#define BETA_Tf 2.0e-2f

#define NBLOCKS 1024
#define NTHREADS 256
// 32 batches * 256 output rows * (256/16) column groups of 16 pixels
#define NUM_TILES (32 * 256 * 16)

// --- Kernel 1: materialize sqrt(alphas_bar[t]) / sqrt(1-alphas_bar[t]) tables ---
__global__ void precompute_schedule(float* __restrict__ sab,
                                    float* __restrict__ somab) {
    if (blockIdx.x == 0 && threadIdx.x == 0) {
        float prod = 1.0f;
        for (int s = 0; s < T_STEPS; ++s) {
            float beta = BETA_1f + (BETA_Tf - BETA_1f) * ((float)s / (float)(T_STEPS - 1));
            prod *= (1.0f - beta);
            sab[s]   = sqrtf(prod);
            somab[s] = sqrtf(1.0f - prod);
        }
    }
}

// --- Kernel 2: fused Haar DWT + forward-diffuse + channel-mix (WMMA) + SSE ---
__global__ __launch_bounds__(NTHREADS) void fused_loss_kernel(
    const float* __restrict__ x0,     // (32, 2, 512, 512)
    const float* __restrict__ noise,  // (32, 4, 256, 256)
    const float* __restrict__ Wm,     // (4, 8)
    const float* __restrict__ bias,   // (4,)
    const float* __restrict__ temb,   // (1000, 4)
    const int*   __restrict__ tidx,   // (32,)
    const float* __restrict__ sab,    // (1000,)
    const float* __restrict__ somab,  // (1000,)
    float* __restrict__ partials)     // (NBLOCKS,)
{
    const int lane = threadIdx.x & 31;
    const int m    = lane & 15;   // A-matrix row / C-matrix column (pixel-in-tile / out-channel)
    const int hi   = lane >> 4;   // half: K pair {0,1} vs {2,3}
    const int k0   = hi << 1;
    const float sgn = hi ? -1.0f : 1.0f;

    const int wavesPerBlock = NTHREADS >> 5;
    const int waveGlobal = blockIdx.x * wavesPerBlock + (threadIdx.x >> 5);
    const int waveStride = NBLOCKS * wavesPerBlock;

    // B tiles (constant over the loop): B[k][n] = W[n][c], c = k (WMMA#1) or 4+k (WMMA#2)
    // Lane holds B[k0+g][m] in element g; columns n>=4 are zero.
    v2f B1 = {0.0f, 0.0f}, B2 = {0.0f, 0.0f};
    if (m < 4) {
        B1.x = Wm[m * 8 + k0 + 0];
        B1.y = Wm[m * 8 + k0 + 1];
        B2.x = Wm[m * 8 + 4 + k0 + 0];
        B2.y = Wm[m * 8 + 4 + k0 + 1];
    }

    float acc = 0.0f;

    for (int tile = waveGlobal; tile < NUM_TILES; tile += waveStride) {
        const int b   = tile >> 12;         // 4096 tiles per batch element
        const int rem = tile & 4095;
        const int i   = rem >> 4;           // output row
        const int j0  = (rem & 15) << 4;    // first of 16 output columns
        const int j   = j0 + m;             // this lane's pixel column

        const int   tb   = tidx[b];
        const float s_ab = sab[tb];
        const float s_om = somab[tb];

        // 2x2 Haar blocks from both image channels (8-byte coalesced loads)
        const float* ct = x0 + (size_t)(b * 2 + 0) * (512 * 512);
        const float* cb = x0 + (size_t)(b * 2 + 1) * (512 * 512);
        const int r0 = i << 1, c0 = j << 1;
        v2f ct_top = *(const v2f*)(ct + (size_t)r0 * 512 + c0);
        v2f ct_bot = *(const v2f*)(ct + (size_t)(r0 + 1) * 512 + c0);
        v2f cb_top = *(const v2f*)(cb + (size_t)r0 * 512 + c0);
        v2f cb_bot = *(const v2f*)(cb + (size_t)(r0 + 1) * 512 + c0);

        // Haar channels k0, k0+1 only:
        //  k=0: (a+b + c+d)/2   k=1: (a+b - c-d)/2
        //  k=2: (a-b + c-d)/2   k=3: (a-b - c+d)/2
        const float pct = ct_top.x + sgn * ct_top.y;
        const float qct = ct_bot.x + sgn * ct_bot.y;
        const float pcb = cb_top.x + sgn * cb_top.y;
        const float qcb = cb_bot.x + sgn * cb_bot.y;
        const v2f ctc = { (pct + qct) * 0.5f, (pct - qct) * 0.5f };
        const v2f cbc = { (pcb + qcb) * 0.5f, (pcb - qcb) * 0.5f };

        // noise channels k0,k0+1 at this pixel (for d_t)
        const float* nbase = noise + (size_t)(b * 4) * 65536 + (size_t)i * 256 + j;
        const float n0 = nbase[(size_t)(k0 + 0) * 65536];
        const float n1 = nbase[(size_t)(k0 + 1) * 65536];

        // A tiles: channels 0-3 = d_t, channels 4-7 = cbct coeffs
        const v2f A1 = { s_ab * (ctc.x - cbc.x) + s_om * n0,
                         s_ab * (ctc.y - cbc.y) + s_om * n1 };
        const v2f A2 = cbc;

        // C accumulator pre-loaded with bias + time embedding (per output column n=m)
        float biasval = 0.0f;
        if (m < 4) biasval = bias[m] + temb[tb * 4 + m];
        v8f C = { biasval, biasval, biasval, biasval,
                  biasval, biasval, biasval, biasval };

        // eps_theta = A1*B1 + A2*B2 + (b + temb) -- two chained f32 WMMAs
        v8f D = __builtin_amdgcn_wmma_f32_16x16x4_f32(false, A1, false, B1,
                                                      (short)0, C, false, false);
        D = __builtin_amdgcn_wmma_f32_16x16x4_f32(false, A2, false, B2,
                                                  (short)0, D, false, false);

        // Squared error: lane (hi, m<4) holds eps for pixels j0+8*hi+r, channel m
        if (m < 4) {
            const float* nd = noise + (size_t)(b * 4 + m) * 65536
                            + (size_t)i * 256 + j0 + hi * 8;
            v4f nA = *(const v4f*)(nd);
            v4f nB = *(const v4f*)(nd + 4);
#pragma unroll
            for (int r = 0; r < 4; ++r) { float d = D[r] - nA[r]; acc += d * d; }
#pragma unroll
            for (int r = 0; r < 4; ++r) { float d = D[4 + r] - nB[r]; acc += d * d; }
        }
    }

    // Deterministic block tree reduction
    __shared__ float red[NTHREADS];
    red[threadIdx.x] = acc;
    __syncthreads();
    for (int s = NTHREADS / 2; s > 0; s >>= 1) {
        if (threadIdx.x < s) red[threadIdx.x] += red[threadIdx.x + s];
        __syncthreads();
    }
    if (threadIdx.x == 0) partials[blockIdx.x] = red[0];
}

// --- Kernel 3: deterministic final reduction of block partials ---
__global__ __launch_bounds__(NTHREADS) void finalize_kernel(
    const float* __restrict__ partials, float* __restrict__ out) {
    __shared__ float red[NTHREADS];
    float s = 0.0f;
    for (int idx = threadIdx.x; idx < NBLOCKS; idx += NTHREADS) s += partials[idx];
    red[threadIdx.x] = s;
    __syncthreads();
    for (int k = NTHREADS / 2; k > 0; k >>= 1) {
        if (threadIdx.x < k) red[threadIdx.x] += red[threadIdx.x + k];
        __syncthreads();
    }
    if (threadIdx.x == 0) out[0] = red[0];
}

extern "C" void kernel_launch(void* const* d_in, const int* in_sizes, int n_in,
                              void* d_out, int out_size, void* d_ws, size_t ws_size,
                              hipStream_t stream) {
    const float* x0    = (const float*)d_in[0];
    const float* noise = (const float*)d_in[1];
    const float* Wm    = (const float*)d_in[2];
    const float* bias  = (const float*)d_in[3];
    const float* temb  = (const float*)d_in[4];
    const int*   tix   = (const int*)d_in[5];

    float* sab      = (float*)d_ws;          // 1000 floats
    float* somab    = sab + T_STEPS;         // 1000 floats
    float* partials = somab + T_STEPS;       // NBLOCKS floats

    precompute_schedule<<<1, 64, 0, stream>>>(sab, somab);
    fused_loss_kernel<<<NBLOCKS, NTHREADS, 0, stream>>>(
        x0, noise, Wm, bias, temb, tix, sab, somab, partials);
    finalize_kernel<<<1, NTHREADS, 0, stream>>>(partials, (float*)d_out);
}